// Decoder_76759655514854
// MI455X (gfx1250) — compile-verified
//
#include <hip/hip_runtime.h>
#include <math.h>

#define BB   8
#define NN   6000
#define CC   21
#define MAXT 200
#define NEGV (-1000000000.0f)
#define THREADS 256
#define KPT  24            // candidates per thread: 256*24 = 6144 >= 6000

typedef unsigned int u32;
typedef __attribute__((ext_vector_type(4))) u32 v4u;
typedef __attribute__((ext_vector_type(8))) int v8i;
typedef __attribute__((ext_vector_type(4))) int v4i;

#if defined(__has_builtin)
#  if __has_builtin(__builtin_amdgcn_tensor_load_to_lds) && __has_builtin(__builtin_amdgcn_s_wait_tensorcnt)
#    define USE_TDM 1
#  endif
#endif

#ifdef USE_TDM
// 1-D contiguous tile copy global->LDS via Tensor Data Mover.
// D# layout per CDNA5 ISA ch.10.8 (group0: count/lds_addr/global_addr/type,
// group1: data_size=4B, tensor_dim0 = tile_dim0 = nElems, stride0 = nElems).
// clang-23 toolchain: 6-arg form (g0 v4u, g1 v8i, g2 v4i, g3 v4i, v8i, cpol).
__device__ __forceinline__ void tdm_load_1d(u32 lds_addr, unsigned long long gaddr, u32 nElems) {
    v4u g0;
    g0[0] = 1u;                                    // count=1, is_restore=0, gather off
    g0[1] = lds_addr;                              // LDS byte address
    g0[2] = (u32)(gaddr & 0xFFFFFFFFull);          // global_addr[31:0]
    g0[3] = (u32)((gaddr >> 32) & 0x1FFFFFFull)    // global_addr[56:32]
          | (2u << 30);                            // type = 2 ("image")
    v8i g1;
    g1[0] = (int)(2u << 16);                       // data_size = 2 (4 bytes), mask=0, no flags
    g1[1] = (int)((nElems & 0xFFFFu) << 16);       // tensor_dim0[15:0] at bits 63:48
    g1[2] = (int)(((nElems >> 16) & 0xFFFFu)       // tensor_dim0[31:16]
          | (1u << 16));                           // tensor_dim1 = 1
    g1[3] = (int)((nElems & 0xFFFFu) << 16);       // tile_dim0 = nElems
    g1[4] = 1;                                     // tile_dim1 = 1, tile_dim2 = 0
    g1[5] = (int)nElems;                           // tensor_dim0_stride lo
    g1[6] = 0;
    g1[7] = 0;
    v4i z4 = {0, 0, 0, 0};
    v8i z8 = {0, 0, 0, 0, 0, 0, 0, 0};
    __builtin_amdgcn_tensor_load_to_lds(g0, g1, z4, z4, z8, 0);
}
#endif

// ---------------------------------------------------------------- phase 2
// one block per (b,c): register-resident greedy NMS; boxes staged in LDS by TDM
// (placed first in the TU so the disasm snippet shows the TDM prologue)
__global__ __launch_bounds__(THREADS)
void nms_kernel(const float* __restrict__ wsBoxes,
                const float* __restrict__ wsScores,
                float* __restrict__ selBox,   // [B][C][MAXT][4]
                float* __restrict__ selVal)   // [B][C][MAXT]
{
    extern __shared__ float smem[];           // [4*NN boxes | NN scores]
    float* sBoxes  = smem;
    float* sScores = smem + 4 * NN;

    __shared__ float redV[THREADS / 32];
    __shared__ int   redI[THREADS / 32];
    __shared__ float winV;
    __shared__ int   winI;

    const int bc  = blockIdx.x;               // b*21 + c
    const int tid = threadIdx.x;
    const float* gBoxes  = wsBoxes  + (size_t)bc * NN * 4;
    const float* gScores = wsScores + (size_t)bc * NN;

#ifdef USE_TDM
    if (tid < 32) {                           // wave 0 issues both DMA descriptors
        tdm_load_1d((u32)(size_t)(void*)sBoxes,
                    (unsigned long long)(size_t)gBoxes, 4u * NN);
        tdm_load_1d((u32)(size_t)(void*)sScores,
                    (unsigned long long)(size_t)gScores, (u32)NN);
        __builtin_amdgcn_s_wait_tensorcnt(0);
    }
#else
    for (int i = tid; i < 4 * NN; i += THREADS) sBoxes[i]  = gBoxes[i];
    for (int i = tid; i < NN;     i += THREADS) sScores[i] = gScores[i];
#endif
    __syncthreads();

    // pull my 24 candidates into registers
    float  sc[KPT];
    float4 bx[KPT];
    float  ar[KPT];
#pragma unroll
    for (int j = 0; j < KPT; ++j) {
        int n = j * THREADS + tid;
        if (n < NN) {
            sc[j] = sScores[n];
            bx[j] = *(const float4*)(sBoxes + 4 * n);
        } else {
            sc[j] = NEGV;
            bx[j] = make_float4(0.f, 0.f, 0.f, 0.f);
        }
        ar[j] = fmaxf(bx[j].z - bx[j].x, 0.0f) * fmaxf(bx[j].w - bx[j].y, 0.0f);
    }

    for (int t = 0; t < MAXT; ++t) {
        // local argmax (value desc, flat index asc)
        float bv = -2.0e9f; int bi = 0x7FFFFFFF;
#pragma unroll
        for (int j = 0; j < KPT; ++j) {
            int n = j * THREADS + tid;
            if (sc[j] > bv || (sc[j] == bv && n < bi)) { bv = sc[j]; bi = n; }
        }
        // wave32 shuffle reduction
#pragma unroll
        for (int off = 16; off > 0; off >>= 1) {
            float ov = __shfl_down(bv, off, 32);
            int   oi = __shfl_down(bi, off, 32);
            if (ov > bv || (ov == bv && oi < bi)) { bv = ov; bi = oi; }
        }
        if ((tid & 31) == 0) { redV[tid >> 5] = bv; redI[tid >> 5] = bi; }
        __syncthreads();
        if (tid == 0) {
            float wv = redV[0]; int wi = redI[0];
#pragma unroll
            for (int w = 1; w < THREADS / 32; ++w) {
                if (redV[w] > wv || (redV[w] == wv && redI[w] < wi)) { wv = redV[w]; wi = redI[w]; }
            }
            winV = wv; winI = wi;
        }
        __syncthreads();

        float val = winV;
        int   nw  = winI;
        float4 wb = *(const float4*)(sBoxes + 4 * nw);   // winner box from LDS
        float  wa = fmaxf(wb.z - wb.x, 0.0f) * fmaxf(wb.w - wb.y, 0.0f);
        bool valid = val > NEGV * 0.5f;

        if (tid == 0) {
            size_t so = (size_t)bc * MAXT + t;
            selVal[so] = valid ? val : NEGV;
            *(float4*)(selBox + so * 4) = wb;
        }

        if (valid) {
#pragma unroll
            for (int j = 0; j < KPT; ++j) {
                float iy1 = fmaxf(wb.x, bx[j].x);
                float ix1 = fmaxf(wb.y, bx[j].y);
                float iy2 = fminf(wb.z, bx[j].z);
                float ix2 = fminf(wb.w, bx[j].w);
                float inter = fmaxf(iy2 - iy1, 0.0f) * fmaxf(ix2 - ix1, 0.0f);
                float iou = inter / (wa + ar[j] - inter + 1e-8f);
                if (iou > 0.5f) sc[j] = NEGV;
            }
        }
    }
}

// ---------------------------------------------------------------- phase 1
// one thread per (b,n): decode+clip 21 boxes, gated thresholded scores
__global__ __launch_bounds__(THREADS)
void decode_kernel(const float* __restrict__ roi,
                   const float* __restrict__ deltas,
                   const float* __restrict__ probs,
                   float* __restrict__ wsBoxes,   // [B][C][N][4]
                   float* __restrict__ wsScores)  // [B][C][N]
{
    int gid = blockIdx.x * THREADS + threadIdx.x;
    if (gid >= BB * NN) return;
    int b = gid / NN;
    int n = gid - b * NN;

    const float4 r = *(const float4*)(roi + (size_t)gid * 4);
    float anc_h  = r.z - r.x;
    float anc_w  = r.w - r.y;
    float anc_cy = r.x + 0.5f * anc_h;
    float anc_cx = r.y + 0.5f * anc_w;

    const float* p = probs + (size_t)gid * CC;
    float pr[CC];
    int lbl = 0; float pm;
#pragma unroll
    for (int c = 0; c < CC; ++c) pr[c] = p[c];
    pm = pr[0];
#pragma unroll
    for (int c = 1; c < CC; ++c) if (pr[c] > pm) { pm = pr[c]; lbl = c; }

    const float* drow = deltas + (size_t)gid * (CC * 4);
#pragma unroll
    for (int c = 0; c < CC; ++c) {
        float4 d = *(const float4*)(drow + c * 4);
        float dy = d.x * 0.1f, dx = d.y * 0.1f;
        float dh = d.z * 0.2f, dw = d.w * 0.2f;
        float bh  = expf(dh) * anc_h;
        float bw  = expf(dw) * anc_w;
        float bcy = dy * anc_h + anc_cy;
        float bcx = dx * anc_w + anc_cx;
        float y1 = bcy - 0.5f * bh;
        float x1 = bcx - 0.5f * bw;
        float y2 = y1 + bh;
        float x2 = x1 + bw;
        y1 = fminf(fmaxf(y1, 0.0f), 1.0f);
        x1 = fminf(fmaxf(x1, 0.0f), 1.0f);
        y2 = fminf(fmaxf(y2, 0.0f), 1.0f);
        x2 = fminf(fmaxf(x2, 0.0f), 1.0f);
        size_t o = (size_t)(b * CC + c) * NN + n;
        float4 ob = make_float4(y1, x1, y2, x2);
        *(float4*)(wsBoxes + o * 4) = ob;
        float s = pr[c];
        wsScores[o] = (lbl != 0 && s > 0.5f) ? s : NEGV;
    }
}

// ---------------------------------------------------------------- phase 3
// one block per batch: iterative top-200 of 4200, gather outputs
__global__ __launch_bounds__(THREADS)
void topk_kernel(const float* __restrict__ selBox,
                 const float* __restrict__ selVal,
                 float* __restrict__ outBox,    // [B][MAXT][4]
                 float* __restrict__ outLbl,    // [B][MAXT]
                 float* __restrict__ outScr)    // [B][MAXT]
{
    const int b   = blockIdx.x;
    const int tid = threadIdx.x;
    const int M   = CC * MAXT;                  // 4200
    __shared__ float sv[CC * MAXT];
    __shared__ float redV[THREADS / 32];
    __shared__ int   redI[THREADS / 32];

    for (int i = tid; i < M; i += THREADS) sv[i] = selVal[(size_t)b * M + i];
    __syncthreads();

    for (int t = 0; t < MAXT; ++t) {
        float bv = -3.0e9f; int bi = 0x7FFFFFFF;
        for (int i = tid; i < M; i += THREADS) {
            float v = sv[i];
            if (v > bv || (v == bv && i < bi)) { bv = v; bi = i; }
        }
#pragma unroll
        for (int off = 16; off > 0; off >>= 1) {
            float ov = __shfl_down(bv, off, 32);
            int   oi = __shfl_down(bi, off, 32);
            if (ov > bv || (ov == bv && oi < bi)) { bv = ov; bi = oi; }
        }
        if ((tid & 31) == 0) { redV[tid >> 5] = bv; redI[tid >> 5] = bi; }
        __syncthreads();
        if (tid == 0) {
            float wv = redV[0]; int wi = redI[0];
#pragma unroll
            for (int w = 1; w < THREADS / 32; ++w) {
                if (redV[w] > wv || (redV[w] == wv && redI[w] < wi)) { wv = redV[w]; wi = redI[w]; }
            }
            bool ok = wv > NEGV * 0.5f;
            float4 fb = *(const float4*)(selBox + ((size_t)b * M + wi) * 4);
            float  fc = (float)(wi / MAXT);
            size_t oo = (size_t)b * MAXT + t;
            float4 zb = make_float4(0.f, 0.f, 0.f, 0.f);
            *(float4*)(outBox + oo * 4) = ok ? fb : zb;
            outLbl[oo] = ok ? fc : 0.0f;
            outScr[oo] = ok ? wv : 0.0f;
            sv[wi] = -2.0e9f;                   // exclude from further picks
        }
        __syncthreads();
    }
}

extern "C" void kernel_launch(void* const* d_in, const int* in_sizes, int n_in,
                              void* d_out, int out_size, void* d_ws, size_t ws_size,
                              hipStream_t stream) {
    (void)in_sizes; (void)n_in; (void)out_size; (void)ws_size;
    const float* roi    = (const float*)d_in[0];   // (B,N,4)
    const float* deltas = (const float*)d_in[1];   // (B,N,C*4)
    const float* probs  = (const float*)d_in[2];   // (B,N,C)

    float* ws = (float*)d_ws;
    float* wsBoxes  = ws;                                    // B*C*N*4
    float* wsScores = wsBoxes  + (size_t)BB * CC * NN * 4;   // B*C*N
    float* selBox   = wsScores + (size_t)BB * CC * NN;       // B*C*MAXT*4
    float* selVal   = selBox   + (size_t)BB * CC * MAXT * 4; // B*C*MAXT

    float* out    = (float*)d_out;
    float* outBox = out;                         // B*MAXT*4 = 6400
    float* outLbl = out + BB * MAXT * 4;         // B*MAXT   = 1600
    float* outScr = outLbl + BB * MAXT;          // B*MAXT   = 1600

    decode_kernel<<<(BB * NN + THREADS - 1) / THREADS, THREADS, 0, stream>>>(
        roi, deltas, probs, wsBoxes, wsScores);

    size_t shmem = (size_t)(4 * NN + NN) * sizeof(float);    // 120 KB / block
    nms_kernel<<<BB * CC, THREADS, shmem, stream>>>(wsBoxes, wsScores, selBox, selVal);

    topk_kernel<<<BB, THREADS, 0, stream>>>(selBox, selVal, outBox, outLbl, outScr);
}